// DeChunkLayer_78915729096798
// MI455X (gfx1250) — compile-verified
//
#include <hip/hip_runtime.h>
#include <stdint.h>

// Problem constants (reference: B=4, L=8192, D=1024)
constexpr int Bb = 4;
constexpr int Ll = 8192;
constexpr int Dd = 1024;
constexpr int T  = 16;          // scan chunk (WMMA M/N)
constexpr int Cc = Ll / T;      // 512 chunks per row
constexpr float EPS = 1e-4f;

typedef float v2f __attribute__((ext_vector_type(2)));
typedef float v8f __attribute__((ext_vector_type(8)));

// ---------------------------------------------------------------------------
// Kernel 1: per-row prep.
//  - stable-partition ranks of boundary tokens (block prefix scan over L)
//  - ps[k] = clipped p at k-th sorted position (zero past Kb -> identity steps)
//  - aa = 1-ps, bpos[k] = linear position of k-th boundary, Pend[c] = prod aa
// ---------------------------------------------------------------------------
__global__ __launch_bounds__(256)
void dechunk_prep(const uint8_t* __restrict__ bm,
                  const float* __restrict__ prob,
                  float* __restrict__ psG, float* __restrict__ aaG,
                  int* __restrict__ bposG, int* __restrict__ KbG,
                  float* __restrict__ PendG)
{
    const int b  = blockIdx.x;
    const int t  = threadIdx.x;
    const int bL = b * Ll;

    __shared__ float ps_s[Ll];     // 32 KB
    __shared__ int   sbuf[256];

    int base = 0;
    for (int tile = 0; tile < Ll / 256; ++tile) {
        const int l = tile * 256 + t;
        const int m = bm[bL + l] ? 1 : 0;

        // Hillis-Steele inclusive prefix sum over 256 lanes
        int x = m;
        sbuf[t] = x;
        __syncthreads();
        for (int off = 1; off < 256; off <<= 1) {
            const int add = (t >= off) ? sbuf[t - off] : 0;
            __syncthreads();
            x += add;
            sbuf[t] = x;
            __syncthreads();
        }
        const int tileTotal = sbuf[255];

        if (m) {
            const int k = base + x - 1;              // stable rank among boundaries
            float p = prob[(size_t)(bL + l) * 2 + 1];
            p = fminf(fmaxf(p, EPS), 1.0f - EPS);
            ps_s[k]        = p;
            bposG[bL + k]  = l;
        }
        base += tileTotal;
        __syncthreads();                             // protect sbuf reuse
    }

    const int Kb = base;                             // #boundaries (>=1: l=0 forced)
    for (int i = Kb + t; i < Ll; i += 256) ps_s[i] = 0.0f;   // identity padding
    __syncthreads();

    for (int i = t; i < Ll; i += 256) {
        const float p = ps_s[i];
        psG[bL + i] = p;
        aaG[bL + i] = 1.0f - p;
    }
    for (int c = t; c < Cc; c += 256) {
        float prod = 1.0f;
#pragma unroll
        for (int j = 0; j < T; ++j) prod *= (1.0f - ps_s[c * T + j]);
        PendG[b * Cc + c] = prod;
    }
    if (t == 0) KbG[b] = Kb;
}

// ---------------------------------------------------------------------------
// Kernel 2: chunk-local scan end value per (b, chunk, channel).
//   h_end = scan of h = a*h + p*x over the 16 steps of the chunk (h init 0)
// ---------------------------------------------------------------------------
__global__ __launch_bounds__(256)
void dechunk_hend(const float* __restrict__ hidden,
                  const float* __restrict__ psG, const float* __restrict__ aaG,
                  const int* __restrict__ KbG, float* __restrict__ hendG)
{
    const int b = blockIdx.z, c = blockIdx.y;
    const int d = blockIdx.x * 256 + threadIdx.x;
    if (c * T >= KbG[b]) return;                     // chunk never observed
    const int bL = b * Ll;

    float h = 0.0f;
#pragma unroll
    for (int j = 0; j < T; ++j) {
        const int k   = c * T + j;
        const float a = aaG[bL + k];
        const float p = psG[bL + k];
        const float x = hidden[(size_t)(bL + k) * Dd + d];
        h = fmaf(a, h, p * x);
    }
    hendG[(size_t)(b * Cc + c) * Dd + d] = h;
}

// ---------------------------------------------------------------------------
// Kernel 3: serial carry combine across chunks (tiny: <=128 live chunks).
//   carry_c = hend_c + Pend_c * carry_{c-1}
// ---------------------------------------------------------------------------
__global__ __launch_bounds__(256)
void dechunk_carry(const float* __restrict__ hendG, const float* __restrict__ PendG,
                   const int* __restrict__ KbG, float* __restrict__ carryG)
{
    const int b = blockIdx.y;
    const int d = blockIdx.x * 256 + threadIdx.x;
    const int Cmax = (KbG[b] + T - 1) / T;
    float prev = 0.0f;
    for (int c = 0; c < Cmax; ++c) {
        const float pe = PendG[b * Cc + c];
        const float he = hendG[(size_t)(b * Cc + c) * Dd + d];
        prev = fmaf(pe, prev, he);
        carryG[(size_t)(b * Cc + c) * Dd + d] = prev;
    }
}

// ---------------------------------------------------------------------------
// Kernel 4: one wave per (b, chunk, 64-column tile).
//   W[k][j] = prod_{i=j+1..k} a_i (16x16 lower-triangular) built in registers,
//   h_local = W @ (p.*x) via 4 chained V_WMMA_F32_16X16X4_F32 per 16-col tile,
//   ema_k = h_local_k + P_k * carry_{c-1}; rows replicated into output
//   segments [bpos[k], bpos[k+1]) with float2 stores (LDS-staged).
// ---------------------------------------------------------------------------
__global__ __launch_bounds__(32)
void dechunk_emit(const float* __restrict__ hidden,
                  const float* __restrict__ psG, const float* __restrict__ aaG,
                  const int* __restrict__ bposG, const int* __restrict__ KbG,
                  const float* __restrict__ carryG, float* __restrict__ out)
{
    const int b       = blockIdx.z;
    const int c       = blockIdx.y;
    const int colBase = blockIdx.x * 64;
    const int lane    = threadIdx.x;
    const int Kb      = KbG[b];
    const int cs      = c * T;
    if (cs >= Kb) return;                            // uniform exit
    const int bL = b * Ll;

    __shared__ float tile[T * 64];                   // 4 KB ema staging

    // CDNA5 prefetch of the hidden rows this wave is about to consume
    __builtin_prefetch(&hidden[(size_t)(bL + cs) * Dd + colBase + lane], 0, 0);

    // Broadcast a/p for the 16 chunk steps
    float aa16[T], ps16[T];
#pragma unroll
    for (int j = 0; j < T; ++j) {
        aa16[j] = aaG[bL + cs + j];
        ps16[j] = psG[bL + cs + j];
    }

    const int  m  = lane & 15;                       // this lane's W row (A-matrix M)
    const bool hi = lane >= 16;

    // Row m of W, descending recurrence: w[m]=1, w[j] = w[j+1]*a[j+1]; 0 above diag.
    float w[T];
    {
        float v = 0.0f;
#pragma unroll
        for (int j = T - 1; j >= 0; --j) {
            v = v * aa16[(j + 1) & 15];              // 0*x stays 0 for j>m
            if (j == m) v = 1.0f;
            w[j] = v;
        }
    }
    // Inclusive prefix products P_k = prod_{i=0..k} a_i (chunk-local)
    float Pv[T];
    {
        float cum = 1.0f;
#pragma unroll
        for (int j = 0; j < T; ++j) { cum *= aa16[j]; Pv[j] = cum; }
    }

    // Four 16-column sub-tiles; each: 4 chained K=4 f32 WMMAs
#pragma unroll
    for (int s = 0; s < 4; ++s) {
        const int col = colBase + 16 * s + m;
        v8f acc = {0.f, 0.f, 0.f, 0.f, 0.f, 0.f, 0.f, 0.f};
#pragma unroll
        for (int q = 0; q < 4; ++q) {
            // A fragment (16x4 f32): lanes0-15 -> K=4q,4q+1 ; lanes16-31 -> K=4q+2,4q+3
            v2f af;
            af.x = hi ? w[4 * q + 2] : w[4 * q + 0];
            af.y = hi ? w[4 * q + 3] : w[4 * q + 1];
            // B fragment (4x16 f32): rows striped across half-waves like C/D
            const float p0 = hi ? ps16[4 * q + 2] : ps16[4 * q + 0];
            const float p1 = hi ? ps16[4 * q + 3] : ps16[4 * q + 1];
            const int   r0 = 4 * q + (hi ? 2 : 0);
            v2f bf;
            bf.x = p0 * hidden[(size_t)(bL + cs + r0)     * Dd + col];
            bf.y = p1 * hidden[(size_t)(bL + cs + r0 + 1) * Dd + col];
            acc = __builtin_amdgcn_wmma_f32_16x16x4_f32(
                /*neg_a=*/false, af, /*neg_b=*/false, bf,
                /*c_mod=*/(short)0, acc, /*reuse_a=*/false, /*reuse_b=*/false);
        }
        // ema_k = h_local_k + P_k * carry_{c-1}[col]; stage to LDS
        const float cprev = (c > 0)
            ? carryG[(size_t)(b * Cc + (c - 1)) * Dd + col] : 0.0f;
#pragma unroll
        for (int r = 0; r < 8; ++r) {
            const float P = hi ? Pv[r + 8] : Pv[r];
            const float e = acc[r] + P * cprev;
            tile[(hi ? r + 8 : r) * 64 + 16 * s + m] = e;
        }
    }

    // Replicate ema rows across their output segments (coalesced float2 stores)
    const int col2 = colBase + 2 * lane;
    for (int k = 0; k < T; ++k) {
        const int kk = cs + k;
        if (kk >= Kb) break;
        const int l0 = bposG[bL + kk];
        const int l1 = (kk + 1 < Kb) ? bposG[bL + kk + 1] : Ll;
        const float2 val = *(const float2*)&tile[k * 64 + 2 * lane];
        for (int l = l0; l < l1; ++l) {
            *(float2*)&out[(size_t)(bL + l) * Dd + col2] = val;
        }
    }
}

// ---------------------------------------------------------------------------
// Launch
// ---------------------------------------------------------------------------
extern "C" void kernel_launch(void* const* d_in, const int* in_sizes, int n_in,
                              void* d_out, int out_size, void* d_ws, size_t ws_size,
                              hipStream_t stream)
{
    (void)in_sizes; (void)n_in; (void)out_size; (void)ws_size;
    const float*   hidden = (const float*)d_in[0];       // (B,L,D) f32
    const uint8_t* bmask  = (const uint8_t*)d_in[1];     // (B,L) bool
    const float*   prob   = (const float*)d_in[2];       // (B,L,2) f32
    /* d_in[3] `mask` unused by the reference */
    float* out = (float*)d_out;

    // Workspace layout (floats): ps | aa | bpos(int) | Kb(int,pad64) | Pend | hend | carry
    float* wsf    = (float*)d_ws;
    float* psG    = wsf;
    float* aaG    = psG + (size_t)Bb * Ll;
    int*   bposG  = (int*)(aaG + (size_t)Bb * Ll);
    int*   KbG    = bposG + (size_t)Bb * Ll;
    float* PendG  = (float*)(KbG + 64);
    float* hendG  = PendG + (size_t)Bb * Cc;
    float* carryG = hendG + (size_t)Bb * Cc * Dd;        // total ~17.2 MB

    dechunk_prep <<<Bb, 256, 0, stream>>>(bmask, prob, psG, aaG, bposG, KbG, PendG);
    dechunk_hend <<<dim3(Dd / 256, Cc, Bb), 256, 0, stream>>>(hidden, psG, aaG, KbG, hendG);
    dechunk_carry<<<dim3(Dd / 256, Bb), 256, 0, stream>>>(hendG, PendG, KbG, carryG);
    dechunk_emit <<<dim3(Dd / 64, Cc, Bb), 32, 0, stream>>>(hidden, psG, aaG, bposG, KbG,
                                                            carryG, out);
}